// GATv2_Layer_71296457114110
// MI455X (gfx1250) — compile-verified
//
#include <hip/hip_runtime.h>
#include <hip/hip_bf16.h>
#include <limits.h>

// ---------------------------------------------------------------------------
// GAT layer, factored form:
//   w = W_attn @ W_fc                      (256 floats)
//   a_s[n] = z[n] . w[0:128]   a_d[n] = z[n] . w[128:256]   (WMMA f32 16x16x4)
//   e[i]  = leaky_relu(a_s[src[i]] + a_d[dst[i]])
//   segment-softmax over dst (atomic max via ordered-int bits, atomic add)
//   h[d] += alpha[i] * z[src[i]]           (float4 loads, f32 atomics, L2-resident)
// ---------------------------------------------------------------------------

#define D 128
#define TWO_D 256

typedef __attribute__((ext_vector_type(2))) float v2f;
typedef __attribute__((ext_vector_type(8))) float v8f;

// order-preserving float<->int mapping for atomicMax on float
__device__ __forceinline__ int f2ord(float f) {
  int i = __float_as_int(f);
  return i >= 0 ? i : (i ^ 0x7FFFFFFF);
}
__device__ __forceinline__ float ord2f(int i) {
  return __int_as_float(i >= 0 ? i : (i ^ 0x7FFFFFFF));
}

// ---- kernel 0: zero output, init per-node max (ordered -inf) and denom -----
__global__ __launch_bounds__(256) void init_kernel(float* __restrict__ out,
                                                   int* __restrict__ m_ord,
                                                   float* __restrict__ denom,
                                                   int n, int total_out) {
  int i = blockIdx.x * blockDim.x + threadIdx.x;
  if (i < total_out) out[i] = 0.0f;
  if (i < n) {
    m_ord[i] = INT_MIN;   // <= f2ord(x) for every float x
    denom[i] = 0.0f;
  }
}

// ---- kernel 1: w[k] = sum_o W_attn[o] * W_fc[o][k]   (one block, 256 thr) --
__global__ __launch_bounds__(256) void attn_w_kernel(const float* __restrict__ Wfc,
                                                     const float* __restrict__ Wattn,
                                                     float* __restrict__ w) {
  int k = threadIdx.x;                 // 0..255
  float acc = 0.0f;
#pragma unroll 8
  for (int o = 0; o < D; ++o)
    acc += Wattn[o] * Wfc[o * TWO_D + k];
  w[k] = acc;
}

// ---- kernel 2: per-node scores via chained V_WMMA_F32_16X16X4_F32 ----------
// One wave handles a 16-node tile. A = z[16 x 4] per K-step (32 steps covers
// K=128), B = 4x16 with column 0 = w_s slice, column 1 = w_d slice, rest 0.
// B is loaded UNCONDITIONALLY from a clamped address (L0-resident 1KB table)
// and masked with a cndmask select — no divergent exec-mask save/restore in
// the inner loop.
// C layout (ISA 7.12.2): lane L holds column N=L&15, VGPR r holds row
// M = r + 8*(L>>4).  So lanes 0/16 hold a_s, lanes 1/17 hold a_d.
__global__ __launch_bounds__(128) void node_scores_kernel(const float* __restrict__ z,
                                                          const float* __restrict__ w,
                                                          float* __restrict__ a_s,
                                                          float* __restrict__ a_d,
                                                          int n) {
  const int lane  = threadIdx.x & 31;
  const int wave  = threadIdx.x >> 5;
  const int tile  = blockIdx.x * 4 + wave;          // 16 nodes per tile
  const int base  = tile * 16;
  if (base >= n) return;                             // wave-uniform exit

  const int ncol  = lane & 15;                       // C column this lane owns
  const int khalf = lane >> 4;                       // A: K pair 0/1 vs 2/3
  int row = base + ncol;                             // A row = node index
  if (row >= n) row = n - 1;                         // safe clamp (N%16==0 normally)

  const float* zrow = z + (size_t)row * D + khalf * 2;

  const bool  bact  = (ncol < 2);
  const int   ncol2 = bact ? ncol : 0;               // clamp -> address valid for all lanes
  const float* wrow = w + ncol2 * D + khalf * 2;     // B[K][N] = w[N*128 + K], N<2

  v8f c = {};
#pragma unroll 4
  for (int kb = 0; kb < 32; ++kb) {                  // K = kb*4 .. kb*4+3
    v2f a;                                           // lane<16 -> K+0,K+1 ; lane>=16 -> K+2,K+3
    a.x = zrow[kb * 4 + 0];
    a.y = zrow[kb * 4 + 1];
    const float bx = wrow[kb * 4 + 0];               // unconditional load
    const float by = wrow[kb * 4 + 1];
    v2f b;
    b.x = bact ? bx : 0.0f;                          // v_cndmask, no branch
    b.y = bact ? by : 0.0f;
    c = __builtin_amdgcn_wmma_f32_16x16x4_f32(false, a, false, b,
                                              (short)0, c, false, false);
  }

  if (ncol == 0) {                                   // column 0 -> a_s
#pragma unroll
    for (int r = 0; r < 8; ++r) {
      int node = base + khalf * 8 + r;
      if (node < n) a_s[node] = c[r];
    }
  } else if (ncol == 1) {                            // column 1 -> a_d
#pragma unroll
    for (int r = 0; r < 8; ++r) {
      int node = base + khalf * 8 + r;
      if (node < n) a_d[node] = c[r];
    }
  }
}

// ---- kernel 3: edge scores + segment max (ordered-int atomicMax) -----------
__global__ __launch_bounds__(256) void edge_score_kernel(const float* __restrict__ a_s,
                                                         const float* __restrict__ a_d,
                                                         const int* __restrict__ src,
                                                         const int* __restrict__ dst,
                                                         float* __restrict__ e,
                                                         int* __restrict__ m_ord,
                                                         int E) {
  int i = blockIdx.x * blockDim.x + threadIdx.x;
  if (i >= E) return;
  int d = dst[i];
  float x = a_s[src[i]] + a_d[d];
  x = x > 0.0f ? x : 0.2f * x;                       // leaky_relu, slope 0.2
  e[i] = x;
  atomicMax(&m_ord[d], f2ord(x));
}

// ---- kernel 4: ex = exp(e - m[dst]); denom[dst] += ex  (e updated in place)
__global__ __launch_bounds__(256) void edge_exp_kernel(const int* __restrict__ dst,
                                                       const int* __restrict__ m_ord,
                                                       float* __restrict__ e,
                                                       float* __restrict__ denom,
                                                       int E) {
  int i = blockIdx.x * blockDim.x + threadIdx.x;
  if (i >= E) return;
  int d = dst[i];
  float m  = ord2f(m_ord[d]);
  float ex = expf(e[i] - m);
  e[i] = ex;
  atomicAdd(&denom[d], ex);
}

// ---- kernel 5: h[dst] += (ex/denom) * z[src]  (one wave per edge) ----------
__global__ __launch_bounds__(256) void aggregate_kernel(const float* __restrict__ z,
                                                        const int* __restrict__ src,
                                                        const int* __restrict__ dst,
                                                        const float* __restrict__ ex,
                                                        const float* __restrict__ denom,
                                                        float* __restrict__ h,
                                                        int E) {
  const int lane = threadIdx.x & 31;
  const int wave = threadIdx.x >> 5;
  const int edge = blockIdx.x * 8 + wave;
  if (edge >= E) return;                             // wave-uniform

  const int s = src[edge];
  const int d = dst[edge];
  const float coeff = ex[edge] / fmaxf(denom[d], 1e-9f);

  const float4 zv = *(const float4*)(z + (size_t)s * D + lane * 4);
  float* hp = h + (size_t)d * D + lane * 4;
  atomicAdd(hp + 0, coeff * zv.x);
  atomicAdd(hp + 1, coeff * zv.y);
  atomicAdd(hp + 2, coeff * zv.z);
  atomicAdd(hp + 3, coeff * zv.w);
}

// ---------------------------------------------------------------------------
extern "C" void kernel_launch(void* const* d_in, const int* in_sizes, int n_in,
                              void* d_out, int out_size, void* d_ws, size_t ws_size,
                              hipStream_t stream) {
  const float* z      = (const float*)d_in[0];   // [N, 128]
  const int*   src    = (const int*)d_in[1];     // [E]
  const int*   dst    = (const int*)d_in[2];     // [E]
  const float* W_fc   = (const float*)d_in[3];   // [128, 256]
  const float* W_attn = (const float*)d_in[4];   // [1, 128]
  float* out = (float*)d_out;                    // [N, 128]

  const int N = in_sizes[0] / D;
  const int E = in_sizes[1];

  // workspace carve-up (~3.2 MB for N=40000, E=640000)
  float* w     = (float*)d_ws;          // 256
  float* a_s   = w + TWO_D;             // N
  float* a_d   = a_s + N;               // N
  int*   m_ord = (int*)(a_d + N);       // N
  float* denom = (float*)(m_ord + N);   // N
  float* e     = denom + N;             // E (reused as ex in place)

  const int total_out = N * D;

  init_kernel<<<(total_out + 255) / 256, 256, 0, stream>>>(out, m_ord, denom, N, total_out);
  attn_w_kernel<<<1, 256, 0, stream>>>(W_fc, W_attn, w);

  const int tiles  = (N + 15) / 16;
  node_scores_kernel<<<(tiles + 3) / 4, 128, 0, stream>>>(z, w, a_s, a_d, N);

  edge_score_kernel<<<(E + 255) / 256, 256, 0, stream>>>(a_s, a_d, src, dst, e, m_ord, E);
  edge_exp_kernel<<<(E + 255) / 256, 256, 0, stream>>>(dst, m_ord, e, denom, E);
  aggregate_kernel<<<(E + 7) / 8, 256, 0, stream>>>(z, src, dst, e, denom, out, E);
}